// RealNet_88184268522216
// MI455X (gfx1250) — compile-verified
//
#include <hip/hip_runtime.h>
#include <math.h>

// ---------------------------------------------------------------------------
// Recurrent cell: h <- LayerNorm(gelu(h @ W + B (+ x_input at t=0)))
// N = 4096, BATCH = 32, STEPS = 64, fp32 throughout.
// GEMM: V_WMMA_F32_16X16X4_F32, each wave computes a 32x32 tile (4 accums)
// so A is loaded once per k-step for both m-tiles and W is read exactly once.
// Split-K = 8 for parallelism; partials reduced inside the fused
// GELU+LayerNorm kernel (vectorized b128 path). Step output slice doubles as
// the next step's hidden state.
// ---------------------------------------------------------------------------

#define NN     4096
#define BATCH  32
#define STEPS  64
#define SPLITK 8
#define KCHUNK (NN / SPLITK)   // 512
#define LN_EPS 1e-5f

typedef float v2f __attribute__((ext_vector_type(2)));
typedef float v4f __attribute__((ext_vector_type(4)));
typedef float v8f __attribute__((ext_vector_type(8)));

// ---------------------------------------------------------------------------
// GEMM partial kernel: part[ks] = h(32 x Kchunk) @ W(Kchunk x 4096) partial.
// grid = (NN/256, SPLITK), block = 256 (8 waves).
// Wave w covers cols [blockIdx.x*256 + w*32, +32) and ALL 32 rows.
// ---------------------------------------------------------------------------
__global__ __launch_bounds__(256)
void rnn_gemm_partial(const float* __restrict__ h,
                      const float* __restrict__ W,
                      float* __restrict__ part)
{
    const int lane = threadIdx.x & 31;
    const int wave = threadIdx.x >> 5;

    const int n0 = blockIdx.x * 256 + wave * 32;   // 32-col wave tile
    const int k0 = blockIdx.y * KCHUNK;            // split-K chunk

    const int r16  = lane & 15;                    // row/col within 16-tile
    const int koff = (lane >> 4) * 2;              // 0 (lanes 0-15) / 2 (16-31)

    // A frags: rows 0-15 (a0) and 16-31 (a1); contiguous float2 per lane
    const float* aptr = h + (size_t)r16 * NN + k0 + koff;
    // B frags: cols n0+r16 (b0) and n0+16+r16 (b1); two rows, stride NN
    const float* bptr = W + (size_t)(k0 + koff) * NN + n0 + r16;

    v8f acc00 = {}, acc01 = {}, acc10 = {}, acc11 = {};

#pragma unroll 2
    for (int k = 0; k < KCHUNK; k += 4) {
        v2f a0 = *(const v2f*)(aptr);
        v2f a1 = *(const v2f*)(aptr + 16 * NN);
        v2f b0, b1;
        b0.x = bptr[0];
        b0.y = bptr[NN];
        b1.x = bptr[16];
        b1.y = bptr[NN + 16];
        acc00 = __builtin_amdgcn_wmma_f32_16x16x4_f32(false, a0, false, b0,
                                                      (short)0, acc00, false, false);
        acc01 = __builtin_amdgcn_wmma_f32_16x16x4_f32(false, a0, false, b1,
                                                      (short)0, acc01, false, false);
        acc10 = __builtin_amdgcn_wmma_f32_16x16x4_f32(false, a1, false, b0,
                                                      (short)0, acc10, false, false);
        acc11 = __builtin_amdgcn_wmma_f32_16x16x4_f32(false, a1, false, b1,
                                                      (short)0, acc11, false, false);
        aptr += 4;
        bptr += 4 * NN;
    }

    // D layout: VGPR r -> row r + (lane>=16 ? 8 : 0) (+16 for acc1x),
    //           col = n0 + r16 (+16 for accx1)
    float* outp = part + (size_t)blockIdx.y * BATCH * NN;
    const int mbase = (lane >> 4) * 8;
    const int c0    = n0 + r16;
#pragma unroll
    for (int r = 0; r < 8; ++r) {
        const size_t rowA = (size_t)(mbase + r) * NN;
        const size_t rowB = (size_t)(16 + mbase + r) * NN;
        outp[rowA + c0]      = acc00[r];
        outp[rowA + c0 + 16] = acc01[r];
        outp[rowB + c0]      = acc10[r];
        outp[rowB + c0 + 16] = acc11[r];
    }
}

// ---------------------------------------------------------------------------
// Fused split-K reduce + bias (+pulse) + exact GELU + LayerNorm.
// grid = BATCH, block = 1024 (32 waves); each thread owns 4 consecutive
// columns so all traffic is b128-vectorized.
// ---------------------------------------------------------------------------
__global__ __launch_bounds__(1024)
void rnn_gelu_layernorm(const float* __restrict__ part,   // [SPLITK][BATCH][NN]
                        const float* __restrict__ bias,   // [NN]
                        const float* __restrict__ gamma,  // [NN]
                        const float* __restrict__ beta,   // [NN]
                        const float* __restrict__ inject, // [BATCH][NN] or null
                        float* __restrict__ out)          // [BATCH][NN]
{
    const int row  = blockIdx.x;
    const int tid  = threadIdx.x;
    const int lane = tid & 31;
    const int wv   = tid >> 5;
    const int c0   = tid * 4;                     // 4 consecutive cols

    __shared__ float red_sum[32];
    __shared__ float red_sq[32];

    const size_t rbase = (size_t)row * NN + c0;

    v4f s = *(const v4f*)(bias + c0);
#pragma unroll
    for (int p = 0; p < SPLITK; ++p)
        s += *(const v4f*)(part + (size_t)p * BATCH * NN + rbase);
    if (inject) s += *(const v4f*)(inject + rbase);

    // exact GELU: 0.5*x*(1+erf(x/sqrt(2)))
    v4f g;
    float sum = 0.0f, sumsq = 0.0f;
#pragma unroll
    for (int i = 0; i < 4; ++i) {
        const float v = 0.5f * s[i] * (1.0f + erff(s[i] * 0.70710678118654752f));
        g[i]  = v;
        sum   += v;
        sumsq += v * v;
    }

    // wave32 reduction, then 32-wave reduction via LDS
#pragma unroll
    for (int off = 16; off > 0; off >>= 1) {
        sum   += __shfl_down(sum,   off, 32);
        sumsq += __shfl_down(sumsq, off, 32);
    }
    if (lane == 0) { red_sum[wv] = sum; red_sq[wv] = sumsq; }
    __syncthreads();

    float ts = 0.0f, tq = 0.0f;
#pragma unroll
    for (int w = 0; w < 32; ++w) { ts += red_sum[w]; tq += red_sq[w]; }

    const float mu   = ts * (1.0f / NN);
    const float var  = tq * (1.0f / NN) - mu * mu;   // population variance
    const float rstd = rsqrtf(var + LN_EPS);

    const v4f ga = *(const v4f*)(gamma + c0);
    const v4f be = *(const v4f*)(beta + c0);
    v4f o;
#pragma unroll
    for (int i = 0; i < 4; ++i)
        o[i] = (g[i] - mu) * rstd * ga[i] + be[i];
    *(v4f*)(out + rbase) = o;
}

// ---------------------------------------------------------------------------
// kernel_launch: 64 steps, 2 kernels per step, all on `stream`.
// d_out layout: [STEPS][BATCH][NN] outputs, then [BATCH][NN] h_final.
// d_ws: SPLITK*BATCH*NN floats (4 MB) of split-K partials.
// ---------------------------------------------------------------------------
extern "C" void kernel_launch(void* const* d_in, const int* in_sizes, int n_in,
                              void* d_out, int out_size, void* d_ws, size_t ws_size,
                              hipStream_t stream)
{
    const float* x_input = (const float*)d_in[0];  // [BATCH][NN]
    const float* h_init  = (const float*)d_in[1];  // [BATCH][NN]
    const float* W       = (const float*)d_in[2];  // [NN][NN]
    const float* B       = (const float*)d_in[3];  // [NN]
    const float* gamma   = (const float*)d_in[4];  // [NN]
    const float* beta    = (const float*)d_in[5];  // [NN]
    (void)in_sizes; (void)n_in; (void)out_size; (void)ws_size;

    float* out  = (float*)d_out;
    float* part = (float*)d_ws;                    // [SPLITK][BATCH][NN]

    const dim3 gemm_grid(NN / 256, SPLITK);
    const dim3 gemm_blk(256);
    const dim3 ln_grid(BATCH);
    const dim3 ln_blk(1024);

    const size_t slice = (size_t)BATCH * NN;

    for (int t = 0; t < STEPS; ++t) {
        const float* h = (t == 0) ? h_init : (out + (size_t)(t - 1) * slice);
        rnn_gemm_partial<<<gemm_grid, gemm_blk, 0, stream>>>(h, W, part);
        rnn_gelu_layernorm<<<ln_grid, ln_blk, 0, stream>>>(
            part, B, gamma, beta,
            (t == 0) ? x_input : nullptr,
            out + (size_t)t * slice);
    }

    // h_final = last step's output slice
    hipMemcpyAsync(out + (size_t)STEPS * slice,
                   out + (size_t)(STEPS - 1) * slice,
                   slice * sizeof(float),
                   hipMemcpyDeviceToDevice, stream);
}